// Block_34832184770611
// MI455X (gfx1250) — compile-verified
//
#include <hip/hip_runtime.h>
#include <hip/hip_bf16.h>
#include <math.h>

#define DI __device__ __forceinline__

typedef __attribute__((ext_vector_type(2))) float v2f;
typedef __attribute__((ext_vector_type(8))) float v8f;

namespace {
constexpr int Bb  = 4;
constexpr int T   = 1024;
constexpr int D   = 1024;
constexpr int H   = 16;
constexpr int Dh  = 64;
constexpr int E   = 8;
constexpr int F   = 4096;
constexpr int N   = Bb * T;        // 4096 tokens
constexpr int CAP = N * 2 / E;     // 1024
}

DI v8f zero8() {
  v8f z;
#pragma unroll
  for (int i = 0; i < 8; ++i) z[i] = 0.f;
  return z;
}

DI v8f wmma4(v2f a, v2f b, v8f c) {
  // D = A(16x4,f32) * B(4x16,f32) + C(16x16,f32)   -> v_wmma_f32_16x16x4_f32
  return __builtin_amdgcn_wmma_f32_16x16x4_f32(false, a, false, b, (short)0, c,
                                               false, false);
}

// Async copy of 16 bytes from global to LDS, tracked by ASYNCcnt.
DI void async_b128(unsigned lds_off, const float* gptr) {
  asm volatile("global_load_async_to_lds_b128 %0, %1, off"
               :: "v"(lds_off), "v"(gptr)
               : "memory");
}
DI void wait_async_le2() { asm volatile("s_wait_asynccnt 0x2" ::: "memory"); }
DI void wait_async_le0() { asm volatile("s_wait_asynccnt 0x0" ::: "memory"); }

// ---------------------------------------------------------------------------
// LayerNorm: one 256-thread block per row of D=1024
// ---------------------------------------------------------------------------
__global__ __launch_bounds__(256) void ln_kernel(const float* __restrict__ x,
                                                 const float* __restrict__ g,
                                                 const float* __restrict__ b,
                                                 float* __restrict__ out) {
  __shared__ float s_sum[256];
  __shared__ float s_sq[256];
  const int row = blockIdx.x;
  const int t   = threadIdx.x;
  const float4 xv = ((const float4*)(x + (size_t)row * D))[t];
  float s = xv.x + xv.y + xv.z + xv.w;
  float q = xv.x * xv.x + xv.y * xv.y + xv.z * xv.z + xv.w * xv.w;
  s_sum[t] = s;
  s_sq[t]  = q;
  __syncthreads();
  for (int off = 128; off > 0; off >>= 1) {
    if (t < off) {
      s_sum[t] += s_sum[t + off];
      s_sq[t]  += s_sq[t + off];
    }
    __syncthreads();
  }
  const float mean = s_sum[0] * (1.f / (float)D);
  const float var  = s_sq[0] * (1.f / (float)D) - mean * mean;
  const float rstd = rsqrtf(var + 1e-5f);
  const float4 gv = ((const float4*)g)[t];
  const float4 bv = ((const float4*)b)[t];
  float4 o;
  o.x = (xv.x - mean) * rstd * gv.x + bv.x;
  o.y = (xv.y - mean) * rstd * gv.y + bv.y;
  o.z = (xv.z - mean) * rstd * gv.z + bv.z;
  o.w = (xv.w - mean) * rstd * gv.w + bv.w;
  ((float4*)(out + (size_t)row * D))[t] = o;
}

// ---------------------------------------------------------------------------
// Generic fp32 WMMA GEMM: C[M x Nc] = op(A[M x K]) @ W[K x Nc]
//   GATHER : A row index indirect through sel[]
//   BIAS   : add bias[n]
//   RELU   : clamp at 0
//   SCATTER: C[sel[m]][n] += gate[m] * value (global_atomic_add_f32)
// Workgroup: 8 waves; block tile 64(M) x 128(N); wave tile 16 x 64.
// B operand double-buffered in LDS via global_load_async_to_lds_b128
// (ASYNCcnt pipeline: prefetch 2 tiles, wait<=2 while one tile in flight).
// ---------------------------------------------------------------------------
template <bool GATHER, bool BIAS, bool RELU, bool SCATTER>
__global__ __launch_bounds__(256) void gemm_wmma_f32(
    const float* __restrict__ A, const float* __restrict__ W,
    const float* __restrict__ bias, float* __restrict__ C, int M, int K, int Nc,
    const int* __restrict__ sel, const float* __restrict__ gate) {
  (void)M;
  constexpr int KT   = 16;    // k-tile depth
  constexpr int ROWP = 132;   // padded row stride in dwords (128 data + 4 pad)
  __shared__ float bsh[2][KT * ROWP];

  const int tid  = threadIdx.x;
  const int lane = tid & 31;
  const int w    = tid >> 5;
  const int l16  = lane & 15;
  const int hi   = lane >> 4;  // 0 or 1
  const int wm   = w & 3;
  const int wn   = w >> 2;
  const int row0 = blockIdx.y * 64 + wm * 16;
  const int col0 = blockIdx.x * 128;
  const int wcol = wn * 64;

  int ar = row0 + l16;
  if (GATHER) ar = sel[ar];
  const float* __restrict__ arow = A + (size_t)ar * K;

  const int ntile = K / KT;
  const unsigned lds_base0 = (unsigned)(size_t)&bsh[0][0];
  const unsigned lds_base1 = (unsigned)(size_t)&bsh[1][0];

  // Issue one 16x128 B-tile: 512 float4, 2 async b128 instructions per thread.
  auto issue = [&](int tile, int buf) {
    const unsigned base = buf ? lds_base1 : lds_base0;
#pragma unroll
    for (int it = 0; it < 2; ++it) {
      const int f    = it * 256 + tid;
      const int krow = f >> 5;   // 0..15
      const int c4   = f & 31;   // 0..31 (float4 column)
      const float* g =
          W + (size_t)(tile * KT + krow) * Nc + col0 + c4 * 4;
      async_b128(base + (unsigned)(krow * ROWP + c4 * 4) * 4u, g);
    }
  };

  issue(0, 0);
  if (ntile > 1) issue(1, 1);

  v8f acc[4];
#pragma unroll
  for (int nt = 0; nt < 4; ++nt) acc[nt] = zero8();

  for (int kt = 0; kt < ntile; ++kt) {
    if (kt == ntile - 1) wait_async_le0();
    else                 wait_async_le2();
    __syncthreads();  // all waves have tile kt resident
    const float* bsp = &bsh[kt & 1][0];
    const int k0 = kt * KT;
#pragma unroll
    for (int k4 = 0; k4 < KT / 4; ++k4) {
      v2f a;
      a.x = arow[k0 + k4 * 4 + hi * 2 + 0];
      a.y = arow[k0 + k4 * 4 + hi * 2 + 1];
      const float* bp = bsp + (k4 * 4 + hi * 2) * ROWP + wcol + l16;
#pragma unroll
      for (int nt = 0; nt < 4; ++nt) {
        v2f bf;
        bf.x = bp[nt * 16];
        bf.y = bp[nt * 16 + ROWP];
        acc[nt] = wmma4(a, bf, acc[nt]);
      }
    }
    __syncthreads();  // everyone done reading buf[kt&1]
    if (kt + 2 < ntile) issue(kt + 2, kt & 1);
  }

#pragma unroll
  for (int nt = 0; nt < 4; ++nt) {
    const int n = col0 + wcol + nt * 16 + l16;
    const float bias_v = BIAS ? bias[n] : 0.f;
#pragma unroll
    for (int r = 0; r < 8; ++r) {
      const int m = row0 + r + hi * 8;
      float v = acc[nt][r] + bias_v;
      if (RELU) v = fmaxf(v, 0.f);
      if (SCATTER) {
        const float gg = gate[m];
        if (gg != 0.f) {
          const int tok = sel[m];
          atomicAdd(C + (size_t)tok * Nc + n, gg * v);
        }
      } else {
        C[(size_t)m * Nc + n] = v;
      }
    }
  }
}

// ---------------------------------------------------------------------------
// RoPE + split qkv (N x 3D, layout [Q|K|V], each (H,Dh)) -> q/k/v (B,H,T,Dh)
// one thread per (token, head, pair j<32)
// ---------------------------------------------------------------------------
__global__ __launch_bounds__(256) void rope_kernel(const float* __restrict__ qkv,
                                                   float* __restrict__ q,
                                                   float* __restrict__ k,
                                                   float* __restrict__ v) {
  const int tid = blockIdx.x * 256 + threadIdx.x;  // N*H*32 total
  const int j = tid & 31;
  const int h = (tid >> 5) & (H - 1);
  const int n = tid >> 9;
  const int b = n >> 10;       // n / T
  const int t = n & (T - 1);   // n % T
  const float inv = __expf(-(float)(2 * j) * (9.2103403719761836f / 64.f));
  const float ang = (float)t * inv;
  const float c = __cosf(ang), s = __sinf(ang);
  const float* row = qkv + (size_t)n * (3 * D);
  const size_t ob = ((size_t)((b * H + h) * T) + t) * Dh;
  const float q1 = row[h * 64 + j], q2 = row[h * 64 + j + 32];
  q[ob + j]      = q1 * c - q2 * s;
  q[ob + j + 32] = q2 * c + q1 * s;
  const float k1 = row[D + h * 64 + j], k2 = row[D + h * 64 + j + 32];
  k[ob + j]      = k1 * c - k2 * s;
  k[ob + j + 32] = k2 * c + k1 * s;
  v[ob + j]      = row[2 * D + h * 64 + j];
  v[ob + j + 32] = row[2 * D + h * 64 + j + 32];
}

// ---------------------------------------------------------------------------
// Flash-style causal attention. One wave per (b,h,16-query tile).
// Scores: 16 chained v_wmma per 16-key tile; P@V: 16 more (P transposed via
// per-wave LDS tile, relying on same-wave in-order DS).
// ctx written as (B,T,D) with col = h*64+dh.
// ---------------------------------------------------------------------------
__global__ __launch_bounds__(256) void attn_kernel(const float* __restrict__ Q,
                                                   const float* __restrict__ Km,
                                                   const float* __restrict__ Vm,
                                                   float* __restrict__ ctx) {
  __shared__ float pbuf[8][16][17];
  const int lane = threadIdx.x & 31;
  const int w    = threadIdx.x >> 5;
  const int l16  = lane & 15;
  const int hi   = lane >> 4;
  const int job  = blockIdx.x * 8 + w;      // B*H*(T/16) = 4096 jobs
  const int qt   = job & 63;
  const int h    = (job >> 6) & (H - 1);
  const int b    = job >> 10;
  const float* qp = Q  + (size_t)((b * H + h) * T) * Dh;
  const float* kp = Km + (size_t)((b * H + h) * T) * Dh;
  const float* vp = Vm + (size_t)((b * H + h) * T) * Dh;

  v2f aq[16];
#pragma unroll
  for (int c = 0; c < 16; ++c) {
    const float* src = qp + (size_t)(qt * 16 + l16) * Dh + c * 4 + hi * 2;
    aq[c].x = src[0];
    aq[c].y = src[1];
  }

  v8f o[4];
#pragma unroll
  for (int nt = 0; nt < 4; ++nt) o[nt] = zero8();
  float mrow[8], lrow[8];
#pragma unroll
  for (int r = 0; r < 8; ++r) {
    mrow[r] = -__builtin_inff();
    lrow[r] = 0.f;
  }

  for (int kt = 0; kt <= qt; ++kt) {
    v8f s = zero8();
#pragma unroll
    for (int c = 0; c < 16; ++c) {
      v2f bk;
      const float* src = kp + (size_t)(kt * 16 + l16) * Dh + c * 4 + hi * 2;
      bk.x = src[0];
      bk.y = src[1];
      s = wmma4(aq[c], bk, s);
    }
#pragma unroll
    for (int r = 0; r < 8; ++r) s[r] *= 0.125f;  // 1/sqrt(64)
    if (kt == qt) {
#pragma unroll
      for (int r = 0; r < 8; ++r)
        if (l16 > r + hi * 8) s[r] = -__builtin_inff();
    }
    float p[8], corr[8];
#pragma unroll
    for (int r = 0; r < 8; ++r) {
      float rm = s[r];
#pragma unroll
      for (int off = 1; off < 16; off <<= 1)
        rm = fmaxf(rm, __shfl_xor(rm, off, 32));
      const float nm = fmaxf(mrow[r], rm);
      const float pe = __expf(s[r] - nm);
      float rs = pe;
#pragma unroll
      for (int off = 1; off < 16; off <<= 1) rs += __shfl_xor(rs, off, 32);
      corr[r] = __expf(mrow[r] - nm);
      lrow[r] = lrow[r] * corr[r] + rs;
      mrow[r] = nm;
      p[r]    = pe;
    }
#pragma unroll
    for (int nt = 0; nt < 4; ++nt)
#pragma unroll
      for (int r = 0; r < 8; ++r) o[nt][r] *= corr[r];

#pragma unroll
    for (int r = 0; r < 8; ++r) pbuf[w][r + hi * 8][l16] = p[r];
    __builtin_amdgcn_sched_barrier(0);
    v2f ap[4];
#pragma unroll
    for (int c2 = 0; c2 < 4; ++c2) {
      ap[c2].x = pbuf[w][l16][c2 * 4 + hi * 2 + 0];
      ap[c2].y = pbuf[w][l16][c2 * 4 + hi * 2 + 1];
    }
#pragma unroll
    for (int c2 = 0; c2 < 4; ++c2) {
#pragma unroll
      for (int nt = 0; nt < 4; ++nt) {
        v2f bv;
        const float* src =
            vp + (size_t)(kt * 16 + c2 * 4 + hi * 2) * Dh + nt * 16 + l16;
        bv.x = src[0];
        bv.y = src[Dh];
        o[nt] = wmma4(ap[c2], bv, o[nt]);
      }
    }
  }

#pragma unroll
  for (int nt = 0; nt < 4; ++nt) {
#pragma unroll
    for (int r = 0; r < 8; ++r) {
      const float ov = o[nt][r] / lrow[r];
      const int tok  = qt * 16 + r + hi * 8;
      ctx[(size_t)(b * T + tok) * D + h * Dh + nt * 16 + l16] = ov;
    }
  }
}

// ---------------------------------------------------------------------------
// elementwise residual: out = a + b   (float4, N*D/4 elements)
// ---------------------------------------------------------------------------
__global__ __launch_bounds__(256) void add_kernel(const float* __restrict__ a,
                                                  const float* __restrict__ b,
                                                  float* __restrict__ out) {
  const int i = blockIdx.x * 256 + threadIdx.x;
  const float4 av = ((const float4*)a)[i];
  const float4 bv = ((const float4*)b)[i];
  float4 o;
  o.x = av.x + bv.x;
  o.y = av.y + bv.y;
  o.z = av.z + bv.z;
  o.w = av.w + bv.w;
  ((float4*)out)[i] = o;
}

// ---------------------------------------------------------------------------
// Router: one wave per token; strided dots + shfl_xor reductions; softplus;
// noisy top-2; 2-way softmax -> routep (N x E).
// ---------------------------------------------------------------------------
__global__ __launch_bounds__(256) void router_kernel(
    const float* __restrict__ h2, const float* __restrict__ noise,
    const float* __restrict__ Wl, const float* __restrict__ bl,
    const float* __restrict__ Wn, const float* __restrict__ bn,
    float* __restrict__ routep) {
  const int lane = threadIdx.x & 31;
  const int w    = threadIdx.x >> 5;
  const int n    = blockIdx.x * 8 + w;
  float aL[E], aN[E];
#pragma unroll
  for (int e = 0; e < E; ++e) { aL[e] = 0.f; aN[e] = 0.f; }
  const float* xr = h2 + (size_t)n * D;
  for (int k = lane; k < D; k += 32) {
    const float xv = xr[k];
    const float* wl = Wl + (size_t)k * E;
    const float* wn = Wn + (size_t)k * E;
#pragma unroll
    for (int e = 0; e < E; ++e) {
      aL[e] = fmaf(xv, wl[e], aL[e]);
      aN[e] = fmaf(xv, wn[e], aN[e]);
    }
  }
#pragma unroll
  for (int e = 0; e < E; ++e) {
#pragma unroll
    for (int off = 16; off > 0; off >>= 1) {
      aL[e] += __shfl_xor(aL[e], off, 32);
      aN[e] += __shfl_xor(aN[e], off, 32);
    }
  }
  float nz[E];
#pragma unroll
  for (int e = 0; e < E; ++e) {
    const float spx = aN[e] + bn[e];
    const float sp  = fmaxf(spx, 0.f) + log1pf(__expf(-fabsf(spx)));
    nz[e] = aL[e] + bl[e] + noise[(size_t)n * E + e] * sp;
  }
  int i1 = 0;
  float v1 = nz[0];
#pragma unroll
  for (int e = 1; e < E; ++e)
    if (nz[e] > v1) { v1 = nz[e]; i1 = e; }
  int i2 = -1;
  float v2 = -__builtin_inff();
#pragma unroll
  for (int e = 0; e < E; ++e)
    if (e != i1 && nz[e] > v2) { v2 = nz[e]; i2 = e; }
  const float e2 = __expf(v2 - v1);
  const float p1 = 1.f / (1.f + e2);
  const float p2 = e2 / (1.f + e2);
  if (lane < E) {
    routep[(size_t)n * E + lane] = (lane == i1) ? p1 : (lane == i2) ? p2 : 0.f;
  }
}

// ---------------------------------------------------------------------------
// Dispatch: one 1024-thread block per expert. Stable in-token-order capacity
// selection via LDS Hillis-Steele scan. Invalid slots -> sel=0, gate=0.
// ---------------------------------------------------------------------------
__global__ __launch_bounds__(1024) void dispatch_kernel(
    const float* __restrict__ routep, int* __restrict__ sel,
    float* __restrict__ gate) {
  __shared__ int scan[1024];
  const int e = blockIdx.x;
  const int t = threadIdx.x;
  float pv[4];
  int msk = 0, cnt = 0;
#pragma unroll
  for (int j = 0; j < 4; ++j) {
    pv[j] = routep[(size_t)(t * 4 + j) * E + e];
    if (pv[j] > 0.f) { msk |= 1 << j; ++cnt; }
  }
  scan[t] = cnt;
  __syncthreads();
  for (int off = 1; off < 1024; off <<= 1) {
    const int add = (t >= off) ? scan[t - off] : 0;
    __syncthreads();
    scan[t] += add;
    __syncthreads();
  }
  const int total = scan[1023];
  int pos = scan[t] - cnt;  // exclusive prefix
#pragma unroll
  for (int j = 0; j < 4; ++j) {
    if ((msk >> j) & 1) {
      if (pos < CAP) {
        sel[e * CAP + pos]  = t * 4 + j;
        gate[e * CAP + pos] = pv[j];
      }
      ++pos;
    }
  }
  const int lim = total < CAP ? total : CAP;
  if (t >= lim) {  // CAP == blockDim.x
    sel[e * CAP + t]  = 0;
    gate[e * CAP + t] = 0.f;
  }
}

// ---------------------------------------------------------------------------
extern "C" void kernel_launch(void* const* d_in, const int* in_sizes, int n_in,
                              void* d_out, int out_size, void* d_ws,
                              size_t ws_size, hipStream_t stream) {
  (void)in_sizes; (void)n_in; (void)out_size; (void)ws_size;
  const float* x     = (const float*)d_in[0];
  const float* noise = (const float*)d_in[1];
  const float* ln1_g = (const float*)d_in[2];
  const float* ln1_b = (const float*)d_in[3];
  const float* ln2_g = (const float*)d_in[4];
  const float* ln2_b = (const float*)d_in[5];
  const float* Wqkv  = (const float*)d_in[6];
  const float* Wproj = (const float*)d_in[7];
  const float* Wr_l  = (const float*)d_in[8];
  const float* br_l  = (const float*)d_in[9];
  const float* Wr_n  = (const float*)d_in[10];
  const float* br_n  = (const float*)d_in[11];
  const float* We1   = (const float*)d_in[12];
  const float* be1   = (const float*)d_in[13];
  const float* We2   = (const float*)d_in[14];
  const float* be2   = (const float*)d_in[15];
  float* out = (float*)d_out;

  float* ws      = (float*)d_ws;
  float* h1      = ws;  ws += (size_t)N * D;
  float* qkv     = ws;  ws += (size_t)N * 3 * D;
  float* qb      = ws;  ws += (size_t)N * D;
  float* kb      = ws;  ws += (size_t)N * D;
  float* vb      = ws;  ws += (size_t)N * D;
  float* ctx     = ws;  ws += (size_t)N * D;
  float* proj    = ws;  ws += (size_t)N * D;
  float* h2      = ws;  ws += (size_t)N * D;
  float* routep  = ws;  ws += (size_t)N * E;
  float* hidden  = ws;  ws += (size_t)CAP * F;
  int*   sel     = (int*)ws;  ws += (size_t)E * CAP;
  float* gate    = ws;

  // 1) h1 = LN1(x)
  ln_kernel<<<N, 256, 0, stream>>>(x, ln1_g, ln1_b, h1);
  // 2) qkv = h1 @ Wqkv   (4096 x 1024 x 3072)
  gemm_wmma_f32<false, false, false, false>
      <<<dim3(3 * D / 128, N / 64), 256, 0, stream>>>(
          h1, Wqkv, nullptr, qkv, N, D, 3 * D, nullptr, nullptr);
  // 3) RoPE + split to (B,H,T,Dh)
  rope_kernel<<<(N * H * 32) / 256, 256, 0, stream>>>(qkv, qb, kb, vb);
  // 4) causal attention -> ctx (B,T,D)
  attn_kernel<<<(Bb * H * (T / 16)) / 8, 256, 0, stream>>>(qb, kb, vb, ctx);
  // 5) proj = ctx @ Wproj
  gemm_wmma_f32<false, false, false, false>
      <<<dim3(D / 128, N / 64), 256, 0, stream>>>(
          ctx, Wproj, nullptr, proj, N, D, D, nullptr, nullptr);
  // 6) out = x + proj   (residual; base for the MoE scatter-add)
  add_kernel<<<(N * D / 4) / 256, 256, 0, stream>>>(x, proj, out);
  // 7) h2 = LN2(out)
  ln_kernel<<<N, 256, 0, stream>>>(out, ln2_g, ln2_b, h2);
  // 8) noisy top-2 routing
  router_kernel<<<N / 8, 256, 0, stream>>>(h2, noise, Wr_l, br_l, Wr_n, br_n,
                                           routep);
  // 9) capacity-limited stable dispatch
  dispatch_kernel<<<E, 1024, 0, stream>>>(routep, sel, gate);
  // 10) experts (serialized on stream; hidden buffer reused)
  for (int e = 0; e < E; ++e) {
    gemm_wmma_f32<true, true, true, false>
        <<<dim3(F / 128, CAP / 64), 256, 0, stream>>>(
            h2, We1 + (size_t)e * D * F, be1 + (size_t)e * F, hidden, CAP, D,
            F, sel + e * CAP, nullptr);
    gemm_wmma_f32<false, true, false, true>
        <<<dim3(D / 128, CAP / 64), 256, 0, stream>>>(
            hidden, We2 + (size_t)e * F * D, be2 + (size_t)e * D, out, CAP, F,
            D, sel + e * CAP, gate + e * CAP);
  }
}